// RadialTokenizer_39376260169803
// MI455X (gfx1250) — compile-verified
//
#include <hip/hip_runtime.h>
#include <hip/hip_bf16.h>

// ---------------------------------------------------------------------------
// RadialTokenizer for MI455X (gfx1250, wave32)
//   Stage 1: masked spatial pooling as a bin-index scatter-reduce (reads the
//            256 MB activation tensor exactly once -> HBM-roofline bound).
//            Deterministic via Q31.32 fixed-point u64 atomics.
//   Stage 2: [4096x128]x[128x768] projection with v_wmma_f32_16x16x4_f32,
//            epilogue fuses bias + positional embedding.
// ---------------------------------------------------------------------------

#define HW      16384      // 128*128
#define C_IN    128
#define T_TOK   128
#define EMBED   768
#define BATCH   32
#define STRIP   32         // pixels per pooling thread
#define NSTRIP  (HW / STRIP)          // 512
#define FIXSCALE 4294967296.0         // 2^32

typedef __attribute__((ext_vector_type(2))) float v2f;
typedef __attribute__((ext_vector_type(8))) float v8f;

// -------- per-pixel bin index from the (disjoint 0/1) masks ----------------
__global__ void rt_idx_kernel(const float* __restrict__ masks,
                              int* __restrict__ idx) {
    int hw = blockIdx.x * blockDim.x + threadIdx.x;   // 16384 threads
    int best = -1;
    for (int t = 0; t < T_TOK; ++t) {
        // coalesced: for fixed t, lanes read consecutive hw
        float m = masks[t * HW + hw];
        if (m != 0.0f && best < 0) best = t;
    }
    idx[hw] = best;
}

// -------- per-bin mask sums -> reciprocal (reference's broadcast divisor) --
__global__ void rt_msum_kernel(const float* __restrict__ masks,
                               float* __restrict__ inv) {
    __shared__ float red[256];
    int t = blockIdx.x;                               // 128 blocks
    float s = 0.0f;
    for (int i = threadIdx.x; i < HW; i += 256)
        s += masks[t * HW + i];
    red[threadIdx.x] = s;
    __syncthreads();
    for (int off = 128; off > 0; off >>= 1) {
        if ((int)threadIdx.x < off) red[threadIdx.x] += red[threadIdx.x + off];
        __syncthreads();
    }
    if (threadIdx.x == 0) inv[t] = 1.0f / red[0];     // inf if empty, like ref
}

// -------- pooling: run-length scatter-reduce into Q31.32 u64 atomics -------
__global__ void rt_pool_kernel(const float* __restrict__ x,
                               const int* __restrict__ idx,
                               unsigned long long* __restrict__ acc64) {
    unsigned int gid = blockIdx.x * blockDim.x + threadIdx.x;  // 2^21 threads
    int strip = gid & (NSTRIP - 1);                   // 9 bits
    int c     = (gid >> 9) & (C_IN - 1);              // 7 bits
    int b     = gid >> 16;                            // 5 bits

    const float4* xp = (const float4*)(x + (((size_t)b * C_IN + c) << 14)
                                         + (size_t)strip * STRIP);
    const int4*   ip = (const int4*)(idx + strip * STRIP);
    unsigned long long* ab = acc64 + ((size_t)b << 14) + c;    // [b][t][c]

    int   cur = -1;
    float run = 0.0f;

    #pragma unroll
    for (int i = 0; i < STRIP / 4; ++i) {
        float4 xv = xp[i];
        int4   iv = ip[i];
        float xs[4] = {xv.x, xv.y, xv.z, xv.w};
        int   ts[4] = {iv.x, iv.y, iv.z, iv.w};
        #pragma unroll
        for (int j = 0; j < 4; ++j) {
            int t = ts[j];
            if (t == cur) {
                run += xs[j];
            } else {
                if (cur >= 0) {
                    long long d = __double2ll_rn((double)run * FIXSCALE);
                    atomicAdd(ab + (size_t)cur * C_IN, (unsigned long long)d);
                }
                cur = t;
                run = xs[j];
            }
        }
    }
    if (cur >= 0) {
        long long d = __double2ll_rn((double)run * FIXSCALE);
        atomicAdd(ab + (size_t)cur * C_IN, (unsigned long long)d);
    }
}

// -------- fixed-point -> float, apply the channel-indexed divisor ----------
__global__ void rt_cvt_kernel(const unsigned long long* __restrict__ acc64,
                              const float* __restrict__ inv,
                              float* __restrict__ pooled) {
    unsigned int i = blockIdx.x * blockDim.x + threadIdx.x;   // 524288 threads
    int c = i & (C_IN - 1);
    long long v = (long long)acc64[i];
    pooled[i] = (float)((double)v * (1.0 / FIXSCALE)) * inv[c];
}

// -------- projection GEMM via fp32 WMMA, fused bias + pos-embed epilogue ---
// D[m,n] = sum_k A[m,k]*B[k,n],  A = pooled [4096 x 128] (K-major),
// B[k,n] = W[n,k]                (W is [768 x 128], also K-major)
__global__ void rt_proj_kernel(const float* __restrict__ pooled,
                               const float* __restrict__ Wp,
                               const float* __restrict__ bias,
                               const float* __restrict__ pos,
                               float* __restrict__ out) {
    const int NT = EMBED / 16;                        // 48 tiles along E
    int wave = threadIdx.x >> 5;
    int lane = threadIdx.x & 31;
    int tile = blockIdx.x * 8 + wave;                 // 12288 tiles total
    int tn = tile % NT;
    int tm = tile / NT;                               // 0..255 (b*T rows /16)

    // 32-bit A 16x4 striping: lane = row + 16*(k>>1), vgpr = k&1.
    // B mirrors with row = n. Both operands are K-major => same address form.
    int row = lane & 15;
    int khi = (lane >> 4) << 1;                       // 0 or 2

    const float* ap = pooled + (size_t)(tm * 16 + row) * C_IN + khi;
    const float* bp = Wp     + (size_t)(tn * 16 + row) * C_IN + khi;

    v8f acc = {};
    #pragma unroll
    for (int k0 = 0; k0 < C_IN; k0 += 4) {
        v2f a = *(const v2f*)(ap + k0);
        v2f b = *(const v2f*)(bp + k0);
        acc = __builtin_amdgcn_wmma_f32_16x16x4_f32(
            /*neg_a=*/false, a, /*neg_b=*/false, b,
            /*c_mod=*/(short)0, acc, /*reuse_a=*/false, /*reuse_b=*/false);
    }

    // C/D layout: VGPR r -> M = r + 8*(lane>=16), N = lane&15
    int n  = lane & 15;
    int mb = (lane >> 4) << 3;
    int ng = tn * 16 + n;
    float be = bias[ng];
    #pragma unroll
    for (int r = 0; r < 8; ++r) {
        int mg = tm * 16 + mb + r;                    // = b*T_TOK + t
        int t  = mg & (T_TOK - 1);
        out[(size_t)mg * EMBED + ng] = acc[r] + be + pos[(size_t)t * EMBED + ng];
    }
}

// ---------------------------------------------------------------------------
extern "C" void kernel_launch(void* const* d_in, const int* in_sizes, int n_in,
                              void* d_out, int out_size, void* d_ws, size_t ws_size,
                              hipStream_t stream) {
    const float* x     = (const float*)d_in[0];   // [32,128,128,128]
    const float* masks = (const float*)d_in[1];   // [128,128,128]
    const float* Wp    = (const float*)d_in[2];   // [768,128]
    const float* bias  = (const float*)d_in[3];   // [768]
    const float* pos   = (const float*)d_in[4];   // [1,128,768]
    float* out = (float*)d_out;                   // [32,128,768]

    const size_t N_POOL = (size_t)BATCH * T_TOK * C_IN;       // 524288
    char* ws = (char*)d_ws;
    unsigned long long* acc64 = (unsigned long long*)ws;       // 4 MB
    float* pooled = (float*)(ws + N_POOL * 8);                 // 2 MB
    int*   idx    = (int*)(ws + N_POOL * 8 + N_POOL * 4);      // 64 KB
    float* inv    = (float*)(ws + N_POOL * 8 + N_POOL * 4 + HW * 4);

    hipMemsetAsync(acc64, 0, N_POOL * 8, stream);

    rt_idx_kernel <<<HW / 256, 256, 0, stream>>>(masks, idx);
    rt_msum_kernel<<<T_TOK,    256, 0, stream>>>(masks, inv);

    // BATCH * C_IN * NSTRIP = 2^21 threads
    rt_pool_kernel<<<(BATCH * C_IN * NSTRIP) / 256, 256, 0, stream>>>(x, idx, acc64);

    rt_cvt_kernel <<<(unsigned)(N_POOL / 256), 256, 0, stream>>>(acc64, inv, pooled);

    // (4096/16)*(768/16) = 12288 tiles, 8 waves (tiles) per 256-thread block
    rt_proj_kernel<<<12288 / 8, 256, 0, stream>>>(pooled, Wp, bias, pos, out);
}